// PointPillarScatter_70970039599805
// MI455X (gfx1250) — compile-verified
//
#include <hip/hip_runtime.h>

// ---------------------------------------------------------------------------
// PointPillarScatter for MI455X (gfx1250).
// Pure data movement (0 FLOPs): scatter [P, 64] pillar features into a dense
// [B, 64, 496, 432] BEV grid. Bandwidth floor at 23.3 TB/s:
//   output stream-write 438.8 MB + feat 32.8 MB + map ~14 MB  ->  ~21 us.
// Strategy: invert scatter -> gather via an L2-resident cell->pillar map so
// every output store is a full 16B lane of a fully-written cacheline
// (no read-modify-write), emitted with TH=NT so the 438 MB stream does not
// evict the hot map/features from the 192 MB L2.
// CDNA5-specific paths: the 6.9 MB map init is done by the Tensor Data Mover
// (tensor_store_from_lds + s_wait_tensorcnt) — DMA fill from a 32 KB LDS tile,
// offloading that store stream from the vector pipes.
// ---------------------------------------------------------------------------

typedef float        f32x4 __attribute__((ext_vector_type(4)));
typedef int          i32x4 __attribute__((ext_vector_type(4)));
typedef unsigned int u32x4 __attribute__((ext_vector_type(4)));
typedef int          i32x8 __attribute__((ext_vector_type(8)));

static constexpr int T_NX  = 432;
static constexpr int T_NY  = 496;
static constexpr int T_C   = 64;
static constexpr int PLANE = T_NX * T_NY;   // 214272 cells per (b, c) plane

// --------------------------- map-based fast path ---------------------------

// TDM map init: block fills a 32KB LDS tile with -1, then wave 0 issues one
// tensor_store_from_lds covering a 32KB chunk of the map. tensor_dim0 clips
// the last partial chunk (writes beyond the tensor are dropped per ISA §8.2).
// Descriptor layout per CDNA5 ISA:
//   group0: [1:0]=count=1, [63:32]=lds_addr, [120:64]=global_addr, [127:126]=2
//   group1: [17:16]=data_size(3 => 8B), [79:48]=tensor_dim0, [111:80]=tensor_dim1,
//           [127:112]=tile_dim0, [143:128]=tile_dim1, [207:160]=dim0_stride
__global__ void PPS_init_map_tdm(int* __restrict__ map,
                                 unsigned long long total_e8) {
    __shared__ int smem[8192];                  // 32 KB DMA source tile
    i32x4* s4 = (i32x4*)smem;
    for (int i = threadIdx.x; i < 2048; i += blockDim.x) {
        i32x4 m = { -1, -1, -1, -1 };
        s4[i] = m;
    }
    __syncthreads();

    if (threadIdx.x == 0) {
        unsigned long long e8_base = (unsigned long long)blockIdx.x * 4096ull;
        unsigned int rem = (unsigned int)(total_e8 - e8_base);
        unsigned int td0 = rem < 4096u ? rem : 4096u;      // clip last chunk
        unsigned long long gaddr =
            (unsigned long long)(uintptr_t)map + e8_base * 8ull;
        unsigned int lds_off = (unsigned int)(uintptr_t)(void*)smem; // low 32b

        u32x4 g0;
        g0.x = 0x1u;                                       // count=1, flags=0
        g0.y = lds_off;                                    // lds_addr
        g0.z = (unsigned int)(gaddr & 0xffffffffull);      // global_addr[31:0]
        g0.w = (unsigned int)(gaddr >> 32) | (2u << 30);   // [56:32] | type=2

        i32x8 g1;
        g1[0] = (3 << 16);                 // data_size=8B; mask/iterate/pad=0
        g1[1] = (int)((td0 & 0xffffu) << 16);              // tensor_dim0[15:0]
        g1[2] = (int)((td0 >> 16) | (1u << 16));           // td0[31:16]|td1=1
        g1[3] = (int)(4096u << 16);                        // tile_dim0 = 4096
        g1[4] = 1;                                         // tile_dim1=1, td2=0
        g1[5] = 4096;                                      // dim0_stride[31:0]
        g1[6] = (int)(4096u << 16);                        // dim1_stride[15:0]
        g1[7] = 0;

        i32x4 gz = { 0, 0, 0, 0 };
#if __clang_major__ >= 23
        i32x8 gz8 = { 0, 0, 0, 0, 0, 0, 0, 0 };
        __builtin_amdgcn_tensor_store_from_lds(g0, g1, gz, gz, gz8, 0);
#else
        __builtin_amdgcn_tensor_store_from_lds(g0, g1, gz, gz, 0);
#endif
        __builtin_amdgcn_s_wait_tensorcnt(0);
    }
}

// map[b*PLANE + z + y*NX + x] = pillar index (indices unique per reference).
__global__ void PPS_build_map(const i32x4* __restrict__ coords,
                              int* __restrict__ map, int P) {
    int p = blockIdx.x * blockDim.x + threadIdx.x;
    if (p < P) {
        i32x4 c = coords[p];                       // (batch, z, y, x) int32
        int cell = c.x * PLANE + c.y + c.z * T_NX + c.w;
        map[cell] = p;
    }
}

// One 128-thread block per (b, y) row; thread t owns x = 4t..4t+3 (t < 108).
// Channel loop in chunks of 4: 4x b128 feature loads (L2 hot), 4x4 register
// transpose, 4x b128 NON-TEMPORAL stores of fully coalesced output rows.
__global__ void PPS_gather(const float* __restrict__ feat,
                           const int* __restrict__ map,
                           float* __restrict__ out) {
    int t = threadIdx.x;
    if (t >= T_NX / 4) return;                     // 108 active lanes
    int row = blockIdx.x;                          // b*NY + y
    int b = row / T_NY;
    int y = row - b * T_NY;

    i32x4 m = *(const i32x4*)(map + (size_t)b * PLANE + (size_t)y * T_NX + 4 * t);
    int p[4] = { m.x, m.y, m.z, m.w };

    float* obase = out + (size_t)b * T_C * PLANE + (size_t)y * T_NX + 4 * t;

    for (int c0 = 0; c0 < T_C; c0 += 4) {
        f32x4 r[4];
#pragma unroll
        for (int i = 0; i < 4; ++i) {
            if (p[i] >= 0) {
                r[i] = *(const f32x4*)(feat + (size_t)p[i] * T_C + c0);
            } else {
                r[i] = (f32x4){ 0.f, 0.f, 0.f, 0.f };
            }
        }
#pragma unroll
        for (int j = 0; j < 4; ++j) {
            f32x4 v = { r[0][j], r[1][j], r[2][j], r[3][j] };
            __builtin_nontemporal_store(v, (f32x4*)(obase + (size_t)(c0 + j) * PLANE));
        }
    }
}

// ------------------------ fallback (no workspace) --------------------------

__global__ void PPS_fill_zero(f32x4* __restrict__ out, int n4) {
    int i = blockIdx.x * blockDim.x + threadIdx.x;
    if (i < n4) {
        f32x4 z = { 0.f, 0.f, 0.f, 0.f };
        __builtin_nontemporal_store(z, &out[i]);
    }
}

// 16 threads per pillar, 4 channels each: b128 feature load, 4 scattered
// dword stores (plane-strided).
__global__ void PPS_scatter(const float* __restrict__ feat,
                            const i32x4* __restrict__ coords,
                            float* __restrict__ out, int P) {
    int tid = blockIdx.x * blockDim.x + threadIdx.x;
    int p = tid >> 4;
    int q = tid & 15;
    if (p >= P) return;
    i32x4 c = coords[p];
    f32x4 f = *(const f32x4*)(feat + (size_t)p * T_C + 4 * q);
    size_t base = (size_t)c.x * T_C * PLANE + (size_t)c.y
                + (size_t)c.z * T_NX + (size_t)c.w;
#pragma unroll
    for (int i = 0; i < 4; ++i)
        out[base + (size_t)(4 * q + i) * PLANE] = f[i];
}

// ------------------------------- launcher ----------------------------------

extern "C" void kernel_launch(void* const* d_in, const int* in_sizes, int n_in,
                              void* d_out, int out_size, void* d_ws, size_t ws_size,
                              hipStream_t stream) {
    const float* feat   = (const float*)d_in[0];
    const i32x4* coords = (const i32x4*)d_in[1];   // int32 (batch, z, y, x)
    const int P = in_sizes[0] / T_C;               // 128000
    const int B = out_size / (T_C * PLANE);        // derive batch from output

    const size_t map_bytes = (size_t)B * PLANE * sizeof(int);

    if (ws_size >= map_bytes) {
        // Fast path: TDM-fill cell->pillar map in d_ws, scatter ids, gather.
        unsigned long long total_e8 = (unsigned long long)(map_bytes / 8);
        unsigned int nchunks = (unsigned int)((total_e8 + 4095ull) / 4096ull);
        PPS_init_map_tdm<<<nchunks, 256, 0, stream>>>((int*)d_ws, total_e8);
        PPS_build_map<<<(P + 255) / 256, 256, 0, stream>>>(coords, (int*)d_ws, P);
        PPS_gather<<<B * T_NY, 128, 0, stream>>>(feat, (const int*)d_ws,
                                                 (float*)d_out);
    } else {
        // Fallback: zero-fill then direct scatter (deterministic in ws_size).
        int n4 = out_size / 4;
        PPS_fill_zero<<<(n4 + 255) / 256, 256, 0, stream>>>((f32x4*)d_out, n4);
        PPS_scatter<<<(P * 16 + 255) / 256, 256, 0, stream>>>(feat, coords,
                                                              (float*)d_out, P);
    }
}